// LightGCN_86861418594408
// MI455X (gfx1250) — compile-verified
//
#include <hip/hip_runtime.h>

#define EMB 64

typedef __attribute__((ext_vector_type(2))) float v2f;
typedef __attribute__((ext_vector_type(8))) float v8f;

// ---------------------------------------------------------------------------
// Guaranteed single-instruction L2 float atomic (no CAS loop).
// ---------------------------------------------------------------------------
__device__ __forceinline__ void atomic_fadd(float* p, float v) {
  asm volatile("global_atomic_add_f32 %0, %1, off scope:SCOPE_DEV"
               :: "v"(p), "v"(v) : "memory");
}

// ---------------------------------------------------------------------------
// 1) Initialize: emb = concat(user_table, item_table); acc (= d_out embs) = emb
// ---------------------------------------------------------------------------
__global__ void lgcn_init(const float* __restrict__ ut, const float* __restrict__ it,
                          float* __restrict__ emb, float* __restrict__ acc,
                          int userFloats, int total4) {
  int i = blockIdx.x * blockDim.x + threadIdx.x;
  if (i >= total4) return;
  int f = i * 4;
  float4 v;
  if (f < userFloats) v = ((const float4*)ut)[i];
  else                v = ((const float4*)it)[(f - userFloats) >> 2];
  ((float4*)emb)[i] = v;
  ((float4*)acc)[i] = v;
}

// ---------------------------------------------------------------------------
// 2) COO scatter SpMM: out[rows[e], :] += vals[e] * emb[cols[e], :]
//    One wave per group of edges; 32 lanes x float2 covers the 64-float row
//    (fully coalesced 256B load per edge; 2 L2 atomics per lane).
// ---------------------------------------------------------------------------
__global__ void lgcn_scatter(const float* __restrict__ emb, float* __restrict__ out,
                             const int* __restrict__ rows, const int* __restrict__ cols,
                             const float* __restrict__ vals, int nnz) {
  const int EPW = 8;  // edges per wave
  int gid  = blockIdx.x * blockDim.x + threadIdx.x;
  int lane = gid & 31;
  long e0  = (long)(gid >> 5) * EPW;
  for (int k = 0; k < EPW; ++k) {
    long e = e0 + k;
    if (e >= nnz) return;
    int   c = cols[e];
    int   r = rows[e];
    float v = vals[e];
    v2f x = ((const v2f*)(emb + (size_t)c * EMB))[lane];
    float* dst = out + (size_t)r * EMB + lane * 2;
    atomic_fadd(dst,     v * x[0]);
    atomic_fadd(dst + 1, v * x[1]);
  }
}

// ---------------------------------------------------------------------------
// 3) acc = (acc + emb) * scale   (scale = 1 for layers 0/1, 0.25 on last)
// ---------------------------------------------------------------------------
__global__ void lgcn_accum(const float* __restrict__ emb, float* __restrict__ acc,
                           float scale, int total4) {
  int i = blockIdx.x * blockDim.x + threadIdx.x;
  if (i >= total4) return;
  float4 a = ((const float4*)acc)[i];
  float4 e = ((const float4*)emb)[i];
  a.x = (a.x + e.x) * scale;
  a.y = (a.y + e.y) * scale;
  a.z = (a.z + e.z) * scale;
  a.w = (a.w + e.w) * scale;
  ((float4*)acc)[i] = a;
}

// ---------------------------------------------------------------------------
// 4) Scoring with f32 WMMA (V_WMMA_F32_16X16X4_F32), full precision.
//    One wave per 16 batch rows. A = user tile (16x4 per step),
//    B = pos/neg item tile (4x16 per step), K = 64 -> 16 WMMA steps per tile.
//    Diagonal of the 16x16 result = the desired dot products.
//    A layout (32-bit 16x4): lane m=l&15; VGPR0/1 hold K = (l<16 ? 0,1 : 2,3).
//    C/D layout: lane n=l&15; VGPR v holds M = v + (l<16 ? 0 : 8).
// ---------------------------------------------------------------------------
__global__ void lgcn_score(const float* __restrict__ F,
                           const int* __restrict__ user, const int* __restrict__ pos,
                           const int* __restrict__ neg,
                           float* __restrict__ pos_out, float* __restrict__ neg_out,
                           int numUsers) {
  __shared__ float tile[256];
  int lane = threadIdx.x;      // 0..31, full wave, EXEC all ones
  int b0   = blockIdx.x * 16;
  int mn   = lane & 15;        // row of A / col of B handled by this lane

  int ui = user[b0 + mn];
  int pi = pos[b0 + mn];
  int ni = neg[b0 + mn];
  const float* uptr = F + (size_t)ui * EMB;
  const float* pptr = F + (size_t)(numUsers + pi) * EMB;
  const float* nptr = F + (size_t)(numUsers + ni) * EMB;

  int khalf = (lane < 16) ? 0 : 2;

  v8f cpos = {0.f, 0.f, 0.f, 0.f, 0.f, 0.f, 0.f, 0.f};
  v8f cneg = {0.f, 0.f, 0.f, 0.f, 0.f, 0.f, 0.f, 0.f};

#pragma unroll
  for (int j = 0; j < 16; ++j) {
    int ka = j * 4 + khalf;
    v2f a  = *(const v2f*)(uptr + ka);
    v2f bp = *(const v2f*)(pptr + ka);
    v2f bn = *(const v2f*)(nptr + ka);
    cpos = __builtin_amdgcn_wmma_f32_16x16x4_f32(false, a, false, bp,
                                                 (short)0, cpos, false, false);
    cneg = __builtin_amdgcn_wmma_f32_16x16x4_f32(false, a, false, bn,
                                                 (short)0, cneg, false, false);
  }

  int mbase = (lane < 16) ? 0 : 8;
#pragma unroll
  for (int v = 0; v < 8; ++v) tile[(mbase + v) * 16 + mn] = cpos[v];
  __syncthreads();
  if (lane < 16) pos_out[b0 + lane] = tile[lane * 17];   // diagonal
  __syncthreads();
#pragma unroll
  for (int v = 0; v < 8; ++v) tile[(mbase + v) * 16 + mn] = cneg[v];
  __syncthreads();
  if (lane < 16) neg_out[b0 + lane] = tile[lane * 17];
}

// ---------------------------------------------------------------------------
// Launch
// ---------------------------------------------------------------------------
extern "C" void kernel_launch(void* const* d_in, const int* in_sizes, int n_in,
                              void* d_out, int out_size, void* d_ws, size_t ws_size,
                              hipStream_t stream) {
  const float* ut   = (const float*)d_in[0];
  const float* it   = (const float*)d_in[1];
  const int*   rows = (const int*)d_in[2];
  const int*   cols = (const int*)d_in[3];
  const float* vals = (const float*)d_in[4];
  const int*   user = (const int*)d_in[5];
  const int*   pos  = (const int*)d_in[6];
  const int*   neg  = (const int*)d_in[7];

  const int userFloats  = in_sizes[0];               // 100000*64
  const int itemFloats  = in_sizes[1];               // 50000*64
  const int totalFloats = userFloats + itemFloats;   // 9,600,000
  const int total4      = totalFloats / 4;
  const int nnz         = in_sizes[2];               // 4,000,000
  const int batch       = in_sizes[5];               // 4096
  const int numUsers    = userFloats / EMB;

  float* embA = (float*)d_ws;                        // ping
  float* embB = embA + totalFloats;                  // pong
  float* pos_out = (float*)d_out;
  float* neg_out = pos_out + batch;
  float* F       = neg_out + batch;                  // final embeddings accumulator

  dim3 blk(256);
  int gridInit = (total4 + 255) / 256;
  lgcn_init<<<gridInit, blk, 0, stream>>>(ut, it, embA, F, userFloats, total4);

  float* src = embA;
  float* dst = embB;
  for (int layer = 0; layer < 3; ++layer) {
    hipMemsetAsync(dst, 0, (size_t)totalFloats * sizeof(float), stream);
    long waves   = (nnz + 7) / 8;          // 8 edges per wave
    long threads = waves * 32;
    int  gridS   = (int)((threads + 255) / 256);
    lgcn_scatter<<<gridS, blk, 0, stream>>>(src, dst, rows, cols, vals, nnz);
    float sc = (layer == 2) ? 0.25f : 1.0f;
    lgcn_accum<<<gridInit, blk, 0, stream>>>(dst, F, sc, total4);
    float* t = src; src = dst; dst = t;
  }

  lgcn_score<<<batch / 16, dim3(32), 0, stream>>>(F, user, pos, neg,
                                                  pos_out, neg_out, numUsers);
}